// LearnableChebyshevKernelAttention_88579405513013
// MI455X (gfx1250) — compile-verified
//
#include <hip/hip_runtime.h>

// ---------------------------------------------------------------------------
// LearnableChebyshevKernelAttention for MI455X (gfx1250, wave32, WMMA)
//
// Stage 1 (fused): per (b,h, 32-row q-block) wave:
//   loop m in chunks of 16:
//     - load 8 key rows/lane ONCE, compute Chebyshev-rational scores for TWO
//       16-row q-tiles directly in WMMA-A f32 register layout
//     - accumulate per-row sum(|s|) for the L1 norm
//     - load each B tile (values) ONCE, feed two v_wmma_f32_16x16x4_f32
//   normalize by 1/(L1+eps), store into flat[B,Q,H*V] layout (workspace).
// Stage 2: out[2048,512] = flat[2048,512] @ out_w^T, 32-row blocks sharing
//   each out_w B tile across two A tiles.
// All matrix math is f32 WMMA (16x16x4) to match the f32 reference exactly.
// ---------------------------------------------------------------------------

typedef __attribute__((ext_vector_type(2))) float v2f;
typedef __attribute__((ext_vector_type(8))) float v8f;

#define WMMA_F32_K4(A, B, C) \
  __builtin_amdgcn_wmma_f32_16x16x4_f32(false, (A), false, (B), (short)0, (C), false, false)

namespace {
constexpr int kB   = 2;
constexpr int kQ   = 1024;
constexpr int kM   = 1024;
constexpr int kH   = 8;
constexpr int kV   = 64;
constexpr int kOUT = 512;
constexpr int kC   = 8;
constexpr int kHV  = kH * kV;   // 512
constexpr float kEPS = 1e-5f;
}

// ---------------------------------------------------------------------------
// Stage 1: fused scores + L1 norm + scores@V
// grid = (Q/32, H, B), block = 32 (one wave, two 16x64 output tiles)
// ---------------------------------------------------------------------------
__global__ __launch_bounds__(32)
void cheb_attn_stage1(const float* __restrict__ qp,          // [B,Q,3]
                      const float* __restrict__ kp,          // [B,M,3]
                      const float* __restrict__ vals,        // [B,M,H,V]
                      const unsigned char* __restrict__ msk, // [B,M] (bool)
                      const float* __restrict__ ls,          // [H]
                      const float* __restrict__ cc,          // [H,C]
                      float* __restrict__ flat)              // [B,Q,H*V]
{
  const int lane = threadIdx.x;       // 0..31
  const int row  = lane & 15;         // column-lane index within a tile
  const int hi   = lane >> 4;         // half-wave: local-K offset 0 or 2
  const int q0   = blockIdx.x * 32;   // 32 query rows per wave (2 tiles)
  const int h    = blockIdx.y;
  const int b    = blockIdx.z;

  // Centered Chebyshev coefficients (force_asymptotic_zero): a[c] -= mean(a)
  float a[kC];
  float mean = 0.f;
#pragma unroll
  for (int c = 0; c < kC; ++c) { a[c] = cc[h * kC + c]; mean += a[c]; }
  mean *= (1.0f / kC);
#pragma unroll
  for (int c = 0; c < kC; ++c) a[c] -= mean;

  const float lsv = ls[h];
  const float inv_ls2 = 1.0f / (lsv * lsv);

  // Query positions for this lane's two rows (tile0: q0+row, tile1: q0+16+row)
  const float* q0p = qp + (size_t)(b * kQ + q0 + row) * 3;
  const float* q1p = qp + (size_t)(b * kQ + q0 + 16 + row) * 3;
  const float qx0 = q0p[0], qy0 = q0p[1], qz0 = q0p[2];
  const float qx1 = q1p[0], qy1 = q1p[1], qz1 = q1p[2];

  v8f acc[2][4] = {};       // two 16x64 attended tiles (4 N-tiles of 16 each)
  float l1a = 0.f, l1b = 0.f;

  for (int m0 = 0; m0 < kM; m0 += 16) {
    // ---- scores in WMMA-A layout; key row shared between both q-tiles
    v2f A0[4], A1[4];
#pragma unroll
    for (int t = 0; t < 4; ++t) {
#pragma unroll
      for (int j = 0; j < 2; ++j) {
        const int m = m0 + 4 * t + 2 * hi + j;
        const float* krow = kp + (size_t)(b * kM + m) * 3;
        const float kx = krow[0], ky = krow[1], kz = krow[2];
        const bool masked = (msk[b * kM + m] != 0);

        // tile 0
        {
          const float dx = qx0 - kx, dy = qy0 - ky, dz = qz0 - kz;
          const float x = (dx * dx + dy * dy + dz * dz) * inv_ls2;
          const float r = (x - 1.0f) / (x + 1.0f);
          float s  = a[0] + a[1] * r;
          float tp = 1.0f, tc = r;
          const float r2 = 2.0f * r;
#pragma unroll
          for (int c = 2; c < kC; ++c) {
            const float tn = __builtin_fmaf(r2, tc, -tp);
            s = __builtin_fmaf(a[c], tn, s);
            tp = tc; tc = tn;
          }
          if (masked) s = 0.f;
          l1a += __builtin_fabsf(s);
          A0[t][j] = s;
        }
        // tile 1 (same key, different query row)
        {
          const float dx = qx1 - kx, dy = qy1 - ky, dz = qz1 - kz;
          const float x = (dx * dx + dy * dy + dz * dz) * inv_ls2;
          const float r = (x - 1.0f) / (x + 1.0f);
          float s  = a[0] + a[1] * r;
          float tp = 1.0f, tc = r;
          const float r2 = 2.0f * r;
#pragma unroll
          for (int c = 2; c < kC; ++c) {
            const float tn = __builtin_fmaf(r2, tc, -tp);
            s = __builtin_fmaf(a[c], tn, s);
            tp = tc; tc = tn;
          }
          if (masked) s = 0.f;
          l1b += __builtin_fabsf(s);
          A1[t][j] = s;
        }
      }
    }
    // ---- B tiles from values[b, m, h, :] loaded once, feed both q-tiles
#pragma unroll
    for (int nt = 0; nt < 4; ++nt) {
      const int v = nt * 16 + row;
#pragma unroll
      for (int t = 0; t < 4; ++t) {
        const int kb = m0 + 4 * t + 2 * hi;   // local K rows {2*hi, 2*hi+1}
        v2f Breg;
        Breg[0] = vals[((size_t)(b * kM + kb    ) * kH + h) * kV + v];
        Breg[1] = vals[((size_t)(b * kM + kb + 1) * kH + h) * kV + v];
        acc[0][nt] = WMMA_F32_K4(A0[t], Breg, acc[0][nt]);
        acc[1][nt] = WMMA_F32_K4(A1[t], Breg, acc[1][nt]);
      }
    }
  }

  // ---- L1 row norms: rows split across lane halves -> xor-16 reduce
  const float rs0 = l1a + __shfl_xor(l1a, 16, 32);
  const float rs1 = l1b + __shfl_xor(l1b, 16, 32);
  __shared__ float inv[32];
  if (lane < 16) {
    inv[lane]      = 1.0f / (rs0 + kEPS);
    inv[lane + 16] = 1.0f / (rs1 + kEPS);
  }
  __syncthreads();

  // ---- scale by 1/L1 and store: C VGPR i holds rows i (lanes<16) / i+8
#pragma unroll
  for (int qt = 0; qt < 2; ++qt) {
#pragma unroll
    for (int nt = 0; nt < 4; ++nt) {
#pragma unroll
      for (int i = 0; i < 8; ++i) {
        const int r = i + 8 * hi;             // row within this 16-row tile
        const float o = acc[qt][nt][i] * inv[qt * 16 + r];
        flat[(size_t)(b * kQ + q0 + qt * 16 + r) * kHV + h * kV + nt * 16 + row] = o;
      }
    }
  }
}

// ---------------------------------------------------------------------------
// Stage 2: out[2048,512] = flat[2048,512] @ out_w^T   (out_w is [512,512])
// grid = (2048/32, 512/64), block = 32; two row-tiles share each B tile.
// ---------------------------------------------------------------------------
__global__ __launch_bounds__(32)
void cheb_attn_outproj(const float* __restrict__ flat,  // [B*Q, 512]
                       const float* __restrict__ w,     // [OUT, 512]
                       float* __restrict__ out)         // [B*Q, OUT]
{
  const int lane = threadIdx.x;
  const int row  = lane & 15;
  const int hi   = lane >> 4;
  const int r0   = blockIdx.x * 32;   // output row block (b*Q+q), 2 tiles
  const int n0   = blockIdx.y * 64;   // output col tile

  v8f acc[2][4] = {};

  for (int k0 = 0; k0 < kHV; k0 += 16) {
    v2f A0[4], A1[4];
#pragma unroll
    for (int t = 0; t < 4; ++t) {
      const int k = k0 + 4 * t + 2 * hi;
      A0[t][0] = flat[(size_t)(r0 + row) * kHV + k];
      A0[t][1] = flat[(size_t)(r0 + row) * kHV + k + 1];
      A1[t][0] = flat[(size_t)(r0 + 16 + row) * kHV + k];
      A1[t][1] = flat[(size_t)(r0 + 16 + row) * kHV + k + 1];
    }
#pragma unroll
    for (int nt = 0; nt < 4; ++nt) {
      const int n = n0 + nt * 16 + row;
#pragma unroll
      for (int t = 0; t < 4; ++t) {
        const int k = k0 + 4 * t + 2 * hi;
        v2f Breg;
        Breg[0] = w[(size_t)n * kHV + k];      // B[k][n] = out_w[n][k]
        Breg[1] = w[(size_t)n * kHV + k + 1];
        acc[0][nt] = WMMA_F32_K4(A0[t], Breg, acc[0][nt]);
        acc[1][nt] = WMMA_F32_K4(A1[t], Breg, acc[1][nt]);
      }
    }
  }

#pragma unroll
  for (int rt = 0; rt < 2; ++rt) {
#pragma unroll
    for (int nt = 0; nt < 4; ++nt) {
#pragma unroll
      for (int i = 0; i < 8; ++i) {
        const int r = i + 8 * hi;
        out[(size_t)(r0 + rt * 16 + r) * kOUT + n0 + nt * 16 + row] = acc[rt][nt][i];
      }
    }
  }
}

// ---------------------------------------------------------------------------
extern "C" void kernel_launch(void* const* d_in, const int* in_sizes, int n_in,
                              void* d_out, int out_size, void* d_ws, size_t ws_size,
                              hipStream_t stream) {
  const float*         qp   = (const float*)d_in[0];          // [B,Q,3]
  const float*         kp   = (const float*)d_in[1];          // [B,M,3]
  const float*         vals = (const float*)d_in[2];          // [B,M,H,V]
  const unsigned char* msk  = (const unsigned char*)d_in[3];  // [B,M] bool
  const float*         ls   = (const float*)d_in[4];          // [H]
  const float*         cc   = (const float*)d_in[5];          // [H,C]
  const float*         w    = (const float*)d_in[6];          // [OUT,H*V]
  float*               out  = (float*)d_out;                  // [B,Q,OUT]
  float*               flat = (float*)d_ws;                   // [B,Q,H*V] scratch (4 MB)

  dim3 g1(kQ / 32, kH, kB);
  cheb_attn_stage1<<<g1, 32, 0, stream>>>(qp, kp, vals, msk, ls, cc, flat);

  dim3 g2((kB * kQ) / 32, kOUT / 64);
  cheb_attn_outproj<<<g2, 32, 0, stream>>>(flat, w, out);
}